// HaarWaveletFrequencyExtractor_28149215658419
// MI455X (gfx1250) — compile-verified
//
#include <hip/hip_runtime.h>
#include <hip/hip_bf16.h>

// ---------------------------------------------------------------------------
// HaarWaveletFrequencyExtractor for MI455X (gfx1250).
//
// Math: Haar filters are orthonormal => x_low[2i+p][2j+q] = 0.25*(a+b+c+d)
// (2x2 block average) and x_high = x - x_low exactly.
//
// Memory-bound op (768 MB traffic, ~33us floor @ 23.3 TB/s). Strategy:
//   * CDNA5 Tensor Data Mover stages contiguous 16 KB input tiles into LDS
//     (one tensor_load_to_lds per tile, issued by wave 0, double-buffered,
//     synchronized with s_wait_tensorcnt + workgroup barriers).
//   * All global stores are non-temporal b128 (streaming 512 MB out; protect
//     the 192 MB L2).
//   * wave32-native 256-thread (8-wave) workgroups, persistent grid-stride.
// ---------------------------------------------------------------------------

typedef float        f32x4 __attribute__((ext_vector_type(4)));
typedef unsigned int u32x4 __attribute__((ext_vector_type(4)));
typedef int          i32x4 __attribute__((ext_vector_type(4)));
typedef int          i32x8 __attribute__((ext_vector_type(8)));

#define WIDTH        512          // fixed by reference shapes [4,64,512,512]
#define TILE_ROWS    8            // rows per TDM tile (contiguous in memory)
#define TILE_ELEMS   (TILE_ROWS * WIDTH)   // 4096 f32 = 16 KB
#define BLOCK        256
#define WPT          2            // (TILE_ELEMS/8)/BLOCK row-pair float4 quads per thread
#define NGRID        2048

#ifndef __has_builtin
#define __has_builtin(x) 0
#endif

#if defined(__AMDGCN__) && __has_builtin(__builtin_amdgcn_tensor_load_to_lds)
#define HAVE_TDM 1
#if __has_include(<hip/amd_detail/amd_gfx1250_TDM.h>)
#define TDM_6ARG 1   // therock-10.0 header present => 6-arg builtin variant
#endif
#endif

// ---- per-row-pair Haar kernel core: two adjacent 2x2 blocks (float4 x 2) ----
__device__ __forceinline__ void haar_emit(f32x4 t, f32x4 u,
                                          float* __restrict__ lo,
                                          float* __restrict__ hi,
                                          size_t b0) {
  const size_t b1 = b0 + WIDTH;
  const float a0 = 0.25f * ((t.x + t.y) + (u.x + u.y));
  const float a1 = 0.25f * ((t.z + t.w) + (u.z + u.w));
  f32x4 low;
  low.x = a0; low.y = a0; low.z = a1; low.w = a1;
  const f32x4 h0 = t - low;
  const f32x4 h1 = u - low;
  // Streaming outputs: non-temporal 128-bit stores.
  __builtin_nontemporal_store(low, (f32x4*)(lo + b0));
  __builtin_nontemporal_store(low, (f32x4*)(lo + b1));
  __builtin_nontemporal_store(h0,  (f32x4*)(hi + b0));
  __builtin_nontemporal_store(h1,  (f32x4*)(hi + b1));
}

#if defined(HAVE_TDM)
// ---- Tensor DMA: 1-D tile of TILE_ELEMS f32 from global into LDS ----------
__device__ __forceinline__ void tdm_issue(const float* __restrict__ x,
                                          size_t elemOff, unsigned ldsByteOff) {
  const unsigned long long ga = (unsigned long long)(const void*)(x + elemOff);
  // D# group 0: count=1 | lds_addr | global_addr[56:0] | type=2
  u32x4 g0;
  g0.x = 1u;                                   // count=1, user descriptor
  g0.y = ldsByteOff;                           // lds_addr (bytes)
  g0.z = (unsigned)ga;                         // global_addr[31:0]
  g0.w = (unsigned)((ga >> 32) & 0x01FFFFFFull) | (2u << 30);  // [56:32] | type=2
  // D# group 1: data_size=4B, tensor_dim0=2^26, tensor_dim1=1,
  //             tile_dim0=TILE_ELEMS (1-D tile: tile_dim1=0 => unused)
  i32x8 g1;
  g1[0] = (int)(2u << 16);                     // workgroup_mask=0, data_size=2 (4B)
  g1[1] = 0;                                   // abar=0, tensor_dim0[15:0]=0
  g1[2] = (int)((0x04000000u >> 16) | (1u << 16)); // dim0[31:16]=0x0400, dim1=1
  g1[3] = (int)((unsigned)TILE_ELEMS << 16);   // dim1[31:16]=0, tile_dim0=4096
  g1[4] = 0;                                   // tile_dim1=0, tile_dim2=0
  g1[5] = (int)0x04000000;                     // tensor_dim0_stride low (unused: 1 line)
  g1[6] = 0;
  g1[7] = 0;
  const i32x4 z4 = {0, 0, 0, 0};
#if defined(TDM_6ARG)
  const i32x8 z8 = {0, 0, 0, 0, 0, 0, 0, 0};
  __builtin_amdgcn_tensor_load_to_lds(g0, g1, z4, z4, z8, 0);
#else
  __builtin_amdgcn_tensor_load_to_lds(g0, g1, z4, z4, 0);
#endif
}

__device__ __forceinline__ void wait_tensorcnt_le1() {
#if __has_builtin(__builtin_amdgcn_s_wait_tensorcnt)
  __builtin_amdgcn_s_wait_tensorcnt(1);
#else
  asm volatile("s_wait_tensorcnt 0x1" ::: "memory");
#endif
}
__device__ __forceinline__ void wait_tensorcnt_le0() {
#if __has_builtin(__builtin_amdgcn_s_wait_tensorcnt)
  __builtin_amdgcn_s_wait_tensorcnt(0);
#else
  asm volatile("s_wait_tensorcnt 0x0" ::: "memory");
#endif
}
#endif  // HAVE_TDM

__global__ __launch_bounds__(BLOCK)
void HaarWaveletFrequencyExtractor_28149215658419_kernel(
    const float* __restrict__ x, float* __restrict__ lo,
    float* __restrict__ hi, int numTiles) {
  const int tid = (int)threadIdx.x;
#if defined(HAVE_TDM)
  __shared__ __align__(16) float lds[2][TILE_ELEMS];   // 2 x 16 KB double buffer
  const bool issuer = (tid < 32);                      // wave 0 drives the TDM
  int tile = (int)blockIdx.x;
  if (tile < numTiles && issuer) {
    tdm_issue(x, (size_t)tile * TILE_ELEMS, (unsigned)(size_t)(void*)&lds[0][0]);
  }
  int buf = 0;
  for (; tile < numTiles; tile += (int)gridDim.x) {
    const int nxt = tile + (int)gridDim.x;
    if (issuer) {
      if (nxt < numTiles) {
        tdm_issue(x, (size_t)nxt * TILE_ELEMS,
                  (unsigned)(size_t)(void*)&lds[buf ^ 1][0]);
        wait_tensorcnt_le1();   // in-order: current tile's DMA has landed
      } else {
        wait_tensorcnt_le0();
      }
    }
    __syncthreads();            // tile data in LDS, visible to all 8 waves

    const size_t tileElem = (size_t)tile * TILE_ELEMS;
#pragma unroll
    for (int s = 0; s < WPT; ++s) {
      const int g  = tid + BLOCK * s;        // 0..511 row-pair quads in tile
      const int rp = g >> 7;                 // row-pair within tile (0..3)
      const int c4 = (g & 127) << 2;         // column (float4-aligned)
      const f32x4 t = *(const f32x4*)&lds[buf][(2 * rp)     * WIDTH + c4];
      const f32x4 u = *(const f32x4*)&lds[buf][(2 * rp + 1) * WIDTH + c4];
      haar_emit(t, u, lo, hi, tileElem + (size_t)(2 * rp) * WIDTH + c4);
    }
    __syncthreads();            // everyone done reading before buf is re-filled
    buf ^= 1;
  }
#else
  // Fallback: direct non-temporal b128 loads (no TDM builtin on this pass).
  for (int tile = (int)blockIdx.x; tile < numTiles; tile += (int)gridDim.x) {
    const size_t tileElem = (size_t)tile * TILE_ELEMS;
#pragma unroll
    for (int s = 0; s < WPT; ++s) {
      const int g  = tid + BLOCK * s;
      const int rp = g >> 7;
      const int c4 = (g & 127) << 2;
      const size_t b0 = tileElem + (size_t)(2 * rp) * WIDTH + c4;
      const f32x4 t = __builtin_nontemporal_load((const f32x4*)(x + b0));
      const f32x4 u = __builtin_nontemporal_load((const f32x4*)(x + b0 + WIDTH));
      haar_emit(t, u, lo, hi, b0);
    }
  }
#endif
}

extern "C" void kernel_launch(void* const* d_in, const int* in_sizes, int n_in,
                              void* d_out, int out_size, void* d_ws, size_t ws_size,
                              hipStream_t stream) {
  (void)n_in; (void)out_size; (void)d_ws; (void)ws_size;
  const float* x  = (const float*)d_in[0];
  const long long N = (long long)in_sizes[0];     // 4*64*512*512 = 2^26
  float* lo = (float*)d_out;                      // x_low flat
  float* hi = lo + N;                             // x_high flat (tuple order)
  const int numTiles = (int)(N / TILE_ELEMS);     // 16384
  const int grid = numTiles < NGRID ? numTiles : NGRID;
  HaarWaveletFrequencyExtractor_28149215658419_kernel<<<grid, BLOCK, 0, stream>>>(
      x, lo, hi, numTiles);
}